// RelationalMatchLSTMEncoder_34866544509166
// MI455X (gfx1250) — compile-verified
//
#include <hip/hip_runtime.h>
#include <math.h>

typedef float v2f __attribute__((ext_vector_type(2)));
typedef float v8f __attribute__((ext_vector_type(8)));
typedef unsigned int u32x4 __attribute__((ext_vector_type(4)));
typedef unsigned int u32x8 __attribute__((ext_vector_type(8)));

#define B_    256
#define H_    512
#define SQ    256
#define SP    256
#define NEGI  (-1000000.0f)

#define PPAD  260   // score row stride (floats): conflict-free phase-2 A reads
#define PC    16    // p-chunk width for phase 2 (double-buffered)
#define PNS   20    // P-chunk LDS row stride: 16 data + 4 pad dwords (TDM pad)

// ---- Tensor Data Mover: 2-D tile load, D# groups 0/1 per ISA 8.3/8.4 ----
// tensor_dim0=256, tensor_dim1=512, tile_dim0=16, tile_dim1=512, stride0=256
__device__ __forceinline__ void tdm_load_2d(unsigned lds_off, const float* gptr,
                                            unsigned g1w0) {
    unsigned long long ga = (unsigned long long)gptr;
    u32x4 g0;
    g0[0] = 1u;                                   // count=1 (valid user D#)
    g0[1] = lds_off;                              // lds_addr (bytes)
    g0[2] = (unsigned)ga;                         // global_addr[31:0]
    g0[3] = ((unsigned)(ga >> 32) & 0x01FFFFFFu)  // global_addr[56:32]
            | 0x80000000u;                        // type=2 ("image")
    u32x8 g1;
    g1[0] = g1w0;                                 // data_size | pad config
    g1[1] = (256u << 16);                         // tensor_dim0[15:0]=256
    g1[2] = (512u << 16);                         // tensor_dim1[15:0]=512
    g1[3] = (16u << 16);                          // tile_dim0=16
    g1[4] = 512u;                                 // tile_dim1=512, tile_dim2=0
    g1[5] = 256u;                                 // tensor_dim0_stride=256
    g1[6] = 0u;
    g1[7] = 0u;
    // 2-D tensor -> 2-group form (VADDR2/3 = NULL)
    asm volatile("tensor_load_to_lds %0, %1" :: "s"(g0), "s"(g1) : "memory");
}

// data_size=2 (4B). P chunks: pad_enable, pad_interval=3 (16 dw), pad_amount=3 (4 dw)
#define G1W0_Q  (2u << 16)
#define G1W0_P  ((2u << 16) | (1u << 20) | (3u << 22) | (3u << 25))

// LDS: lds_q 32KB + lds_s 16.6KB + lds_pn 80KB + masks ~1.1KB ~= 132KB -> 2 blk/WGP
__global__ __launch_bounds__(256)
void match_attn_kernel(const float* __restrict__ qh,
                       const float* __restrict__ ph,
                       const int*   __restrict__ qm,
                       const int*   __restrict__ pm,
                       float*       __restrict__ out)
{
    __shared__ float lds_q[H_ * 16];        // Q tile [h][ql], stride 16 (TDM, no pad)
    __shared__ float lds_s[16 * PPAD];      // scores/att [q][p]
    __shared__ float lds_pn[2][H_ * PNS];   // P chunk [h][p_local], stride 20 (TDM pad)
    __shared__ int   lds_pm[SP];
    __shared__ int   lds_qm[16];

    const int bidx = blockIdx.y;
    const int q0   = blockIdx.x * 16;
    const int t    = threadIdx.x;
    const int lane = t & 31;
    const int wave = t >> 5;
    const int ln   = lane & 15;             // lane % 16
    const int krow = (lane >> 4) << 1;      // 2*(lane/16)

    const float* Q = qh + (size_t)bidx * H_ * SQ;
    const float* P = ph + (size_t)bidx * H_ * SP;

    // ---------- stage Q tile via TDM; prefetch P chunk 0; load masks ----------
    if (wave == 0) {
        tdm_load_2d((unsigned)(size_t)&lds_q[0],     Q + q0, G1W0_Q);
        tdm_load_2d((unsigned)(size_t)&lds_pn[0][0], P,      G1W0_P);
    }
    if (t < 16) lds_qm[t] = qm[bidx * SQ + q0 + t];
    lds_pm[t] = pm[bidx * SP + t];
    if (wave == 0) __builtin_amdgcn_s_wait_tensorcnt(1);   // Q tile landed
    __syncthreads();

    // ---------- phase 1: sim tile (16q x 32p per wave), K over H ----------
    {
        const int p0 = wave * 32;
        v8f acc0 = {}; v8f acc1 = {};
        const float* Pb = P + (size_t)krow * SP + p0 + ln;
        #pragma unroll 4
        for (int k = 0; k < H_; k += 4) {
            v2f a, b0, b1;
            a.x = lds_q[(k + krow)     * 16 + ln];
            a.y = lds_q[(k + krow + 1) * 16 + ln];
            const float* pk = Pb + (size_t)k * SP;
            b0.x = pk[0];       b0.y = pk[SP];
            b1.x = pk[16];      b1.y = pk[SP + 16];
            acc0 = __builtin_amdgcn_wmma_f32_16x16x4_f32(false, a, false, b0,
                                                         (short)0, acc0, false, false);
            acc1 = __builtin_amdgcn_wmma_f32_16x16x4_f32(false, a, false, b1,
                                                         (short)0, acc1, false, false);
        }
        const int qrow = (lane >> 4) * 8;
        #pragma unroll
        for (int r = 0; r < 8; ++r) {
            lds_s[(qrow + r) * PPAD + p0 + ln]      = acc0[r];
            lds_s[(qrow + r) * PPAD + p0 + 16 + ln] = acc1[r];
        }
    }
    __syncthreads();

    // ---------- masked softmax over p: wave w owns rows 2w, 2w+1 ----------
    {
        #pragma unroll
        for (int rr = 0; rr < 2; ++rr) {
            const int q = wave * 2 + rr;
            const float qmv = (float)lds_qm[q];
            float v[8];
            float mx = -1e30f;
            #pragma unroll
            for (int j = 0; j < 8; ++j) {
                const int p = lane + 32 * j;
                float s = lds_s[q * PPAD + p] + qmv * (float)lds_pm[p] * NEGI;
                v[j] = s;
                mx = fmaxf(mx, s);
            }
            #pragma unroll
            for (int off = 16; off > 0; off >>= 1)
                mx = fmaxf(mx, __shfl_xor(mx, off, 32));
            float sum = 0.0f;
            #pragma unroll
            for (int j = 0; j < 8; ++j) {
                v[j] = __expf(v[j] - mx);
                sum += v[j];
            }
            #pragma unroll
            for (int off = 16; off > 0; off >>= 1)
                sum += __shfl_xor(sum, off, 32);
            const float inv = 1.0f / sum;
            #pragma unroll
            for (int j = 0; j < 8; ++j)
                lds_s[q * PPAD + lane + 32 * j] = v[j] * inv;
        }
    }
    __syncthreads();

    // ---------- phase 2: q_tilda = att x P^T, TDM double-buffered ----------
    v8f tacc[4] = {v8f{}, v8f{}, v8f{}, v8f{}};
    for (int c = 0; c < SP / PC; ++c) {
        if (wave == 0) __builtin_amdgcn_s_wait_tensorcnt(0);  // chunk c landed
        __syncthreads();                                      // visible to all waves
        if (wave == 0 && c + 1 < SP / PC)                     // overlap DMA c+1
            tdm_load_2d((unsigned)(size_t)&lds_pn[(c + 1) & 1][0],
                        P + (c + 1) * PC, G1W0_P);
        const float* pn = &lds_pn[c & 1][0];
        const int pc = c * PC;
        #pragma unroll
        for (int kk = 0; kk < PC; kk += 4) {
            v2f a;
            a.x = lds_s[ln * PPAD + pc + kk + krow];
            a.y = lds_s[ln * PPAD + pc + kk + krow + 1];
            #pragma unroll
            for (int tl = 0; tl < 4; ++tl) {
                const int h0 = wave * 16 + tl * 128;
                v2f bb;
                bb.x = pn[(h0 + ln) * PNS + kk + krow];
                bb.y = pn[(h0 + ln) * PNS + kk + krow + 1];
                tacc[tl] = __builtin_amdgcn_wmma_f32_16x16x4_f32(false, a, false, bb,
                                                                 (short)0, tacc[tl],
                                                                 false, false);
            }
        }
        __syncthreads();   // compute on this buffer done before TDM reuses it
    }

    // ---------- epilogue: fused 4-way concat [h, h~, h-h~, h*h~] ----------
    {
        const size_t ob   = ((size_t)bidx * SQ + q0) * (size_t)(4 * H_);
        const int    qrow = (lane >> 4) * 8;
        #pragma unroll
        for (int tl = 0; tl < 4; ++tl) {
            const int h = wave * 16 + tl * 128 + ln;
            #pragma unroll
            for (int r = 0; r < 8; ++r) {
                const int q   = qrow + r;
                const float til = tacc[tl][r];
                const float qt  = lds_q[h * 16 + q];
                float* o = out + ob + (size_t)q * (4 * H_);
                o[h]          = qt;
                o[H_ + h]     = til;
                o[2 * H_ + h] = qt - til;
                o[3 * H_ + h] = qt * til;
            }
        }
    }
}

extern "C" void kernel_launch(void* const* d_in, const int* in_sizes, int n_in,
                              void* d_out, int out_size, void* d_ws, size_t ws_size,
                              hipStream_t stream) {
    (void)in_sizes; (void)n_in; (void)d_ws; (void)ws_size; (void)out_size;
    const float* q_hiddens = (const float*)d_in[0];
    const float* p_hiddens = (const float*)d_in[1];
    const int*   q_mask    = (const int*)d_in[2];
    const int*   p_mask    = (const int*)d_in[3];
    float*       out       = (float*)d_out;

    dim3 grid(SQ / 16, B_);   // 16 q-tiles x 256 batches
    dim3 block(256);          // 8 wave32
    match_attn_kernel<<<grid, block, 0, stream>>>(q_hiddens, p_hiddens,
                                                  q_mask, p_mask, out);
}